// MFA_block_747324309591
// MI455X (gfx1250) — compile-verified
//
#include <hip/hip_runtime.h>
#include <math.h>

#define DEVFN __device__ __forceinline__

typedef float v2f __attribute__((ext_vector_type(2)));
typedef float v8f __attribute__((ext_vector_type(8)));

struct alignas(16) f4 {
  float x, y, z, w;
};

DEVFN float gelu_erf(float x) {
  return 0.5f * x * (1.0f + erff(x * 0.7071067811865475f));
}

// D = A(16x4) * B(4x16) + C, fp32 WMMA (CDNA5 gfx1250)
DEVFN v8f wmma_f32_16x16x4(v2f a, v2f b, v8f c) {
  return __builtin_amdgcn_wmma_f32_16x16x4_f32(false, a, false, b, (short)0, c,
                                               false, false);
}

// ---------------- BN scale/shift prep: sc = g/sqrt(v+eps), sh = b - m*sc ----
__global__ void bn_prep_kernel(const float* __restrict__ g,
                               const float* __restrict__ b,
                               const float* __restrict__ m,
                               const float* __restrict__ v,
                               float* __restrict__ sc, float* __restrict__ sh,
                               int C) {
  int i = blockIdx.x * blockDim.x + threadIdx.x;
  if (i < C) {
    float s = g[i] / sqrtf(v[i] + 1e-5f);
    sc[i] = s;
    sh[i] = b[i] - m[i] * s;
  }
}

// ---------------- depthwise 3x3 (SAME, zero pad), W must be 32 --------------
// MODE 0: out = center + (conv + bias)                      (LPU)
// MODE 1: out = center + gelu((conv + bias)*sc + sh)        (IRFFN h2)
template <int MODE>
__global__ __launch_bounds__(256) void dwconv3x3_kernel(
    const float* __restrict__ in, const float* __restrict__ wgt,
    const float* __restrict__ bias, const float* __restrict__ sc,
    const float* __restrict__ sh, float* __restrict__ out, int C, int H,
    int W) {
  int x = threadIdx.x & 31;   // W == 32
  int ry = threadIdx.x >> 5;  // 0..7
  int h = blockIdx.x * 8 + ry;
  int c = blockIdx.y;
  int b = blockIdx.z;
  long base = ((long)b * C + c) * (long)(H * W);
  const float* wp = wgt + c * 9;
  float acc = bias[c];
#pragma unroll
  for (int dy = -1; dy <= 1; ++dy) {
    int hh = h + dy;
    if (hh < 0 || hh >= H) continue;
#pragma unroll
    for (int dx = -1; dx <= 1; ++dx) {
      int ww = x + dx;
      if (ww < 0 || ww >= W) continue;
      acc += wp[(dy + 1) * 3 + (dx + 1)] * in[base + (long)hh * W + ww];
    }
  }
  float center = in[base + (long)h * W + x];
  float r;
  if (MODE == 0) {
    r = center + acc;
  } else {
    float t = acc * sc[c] + sh[c];
    r = center + gelu_erf(t);
  }
  out[base + (long)h * W + x] = r;
}

// ---------------- LayerNorm over channel dim of NCHW ----------------
__global__ __launch_bounds__(256) void ln_chan_kernel(
    const float* __restrict__ in, const float* __restrict__ g,
    const float* __restrict__ bb, float* __restrict__ out, int C, int N) {
  long idx = (long)blockIdx.x * blockDim.x + threadIdx.x;
  int b = (int)(idx / N);
  int n = (int)(idx % N);
  long base = ((long)b * C) * (long)N + n;
  float s = 0.f, s2 = 0.f;
  for (int c = 0; c < C; ++c) {
    float x = in[base + (long)c * N];
    s += x;
    s2 += x * x;
  }
  float mu = s / (float)C;
  float var = s2 / (float)C - mu * mu;
  var = fmaxf(var, 0.f);
  float rs = 1.0f / sqrtf(var + 1e-5f);
  for (int c = 0; c < C; ++c) {
    float x = in[base + (long)c * N];
    out[base + (long)c * N] = (x - mu) * rs * g[c] + bb[c];
  }
}

// ---------------- E[b][64][64] = sum_n A[b][64][N] * B[b][64][N]^T ----------
__global__ __launch_bounds__(256) void abt64_kernel(
    const float* __restrict__ A, const float* __restrict__ Bt,
    float* __restrict__ E, int N, long aBS, long bBS) {
  __shared__ float As[64][65];
  __shared__ float Bs[64][65];
  int b = blockIdx.x;
  const float* Ab = A + (long)b * aBS;
  const float* Bb = Bt + (long)b * bBS;
  int tid = threadIdx.x;
  int c = tid & 63;
  int dbase = (tid >> 6) * 16;
  int q = tid & 15, r0 = tid >> 4;  // staging: 16 quads x 16 rows (x4)
  float acc[16];
#pragma unroll
  for (int i = 0; i < 16; ++i) acc[i] = 0.f;
  for (int n0 = 0; n0 < N; n0 += 64) {
    f4 ra[4], rb[4];
#pragma unroll
    for (int i = 0; i < 4; ++i) {
      long o = (long)(r0 + 16 * i) * N + n0 + q * 4;
      ra[i] = *(const f4*)&Ab[o];
      rb[i] = *(const f4*)&Bb[o];
    }
#pragma unroll
    for (int i = 0; i < 4; ++i) {
      int r = r0 + 16 * i;
      As[r][q * 4 + 0] = ra[i].x;
      As[r][q * 4 + 1] = ra[i].y;
      As[r][q * 4 + 2] = ra[i].z;
      As[r][q * 4 + 3] = ra[i].w;
      Bs[r][q * 4 + 0] = rb[i].x;
      Bs[r][q * 4 + 1] = rb[i].y;
      Bs[r][q * 4 + 2] = rb[i].z;
      Bs[r][q * 4 + 3] = rb[i].w;
    }
    __syncthreads();
#pragma unroll 4
    for (int n = 0; n < 64; ++n) {
      float av = As[c][n];
#pragma unroll
      for (int i = 0; i < 16; ++i) acc[i] += av * Bs[dbase + i][n];
    }
    __syncthreads();
  }
#pragma unroll
  for (int i = 0; i < 16; ++i)
    E[(long)b * 4096 + (long)c * 64 + dbase + i] = acc[i];
}

// ---------------- M2[b][256][64] = (Wc[256,64] @ E[b][64,64]) / 64 ----------
__global__ __launch_bounds__(256) void fold_cnl_kernel(
    const float* __restrict__ Wc, const float* __restrict__ E,
    float* __restrict__ M2) {
  int idx = blockIdx.x * 256 + threadIdx.x;  // [0, 16384)
  int b = blockIdx.y;
  int o = idx >> 6, m = idx & 63;
  const float* Eb = E + (long)b * 4096;
  float acc = 0.f;
#pragma unroll 8
  for (int e = 0; e < 64; ++e) acc += Wc[o * 64 + e] * Eb[e * 64 + m];
  M2[(long)b * 16384 + o * 64 + m] = acc * (1.0f / 64.0f);
}

// -------- Wf[b][half][256][64]: fold pnl_W through S (reshape-aware) --------
// Wf[half][o][m] = (1/1536) * sum_{k<32} Wp[o][k] * S[m][2k+half]
__global__ __launch_bounds__(256) void fold_pnl_kernel(
    const float* __restrict__ Wp, const float* __restrict__ S,
    float* __restrict__ Wf) {
  int idx = blockIdx.x * 256 + threadIdx.x;  // [0, 32768)
  int b = blockIdx.y;
  int half = idx >> 14;
  int o = (idx >> 6) & 255;
  int m = idx & 63;
  const float* Sb = S + (long)b * 4096;
  float acc = 0.f;
#pragma unroll 8
  for (int k = 0; k < 32; ++k)
    acc += Wp[o * 32 + k] * Sb[m * 64 + 2 * k + half];
  Wf[(((long)b * 2 + half) * 256 + o) * 64 + m] = acc * (1.0f / 1536.0f);
}

// ---------------- batched WMMA fp32 GEMM: Out[bz] = epi(A @ B) --------------
// Block tile 64(M) x 128(N); 8 waves: 4 M-waves x 2 N-waves, wave tile 16x64.
// Software-pipelined K-chunks of 32 with b128 global->reg->LDS staging.
enum { EPI_BIAS = 0, EPI_BN_RES = 1, EPI_GELU_BN = 2 };

template <int EPI>
__global__ __launch_bounds__(256) void gemm_wmma_kernel(
    const float* __restrict__ A, long aBS, const float* __restrict__ Bm,
    long bBS, int ldb, float* __restrict__ Out, long oBS, int ldo,
    long oHalfOff, int halfShift, const float* __restrict__ bias,
    const float* __restrict__ bnsc, const float* __restrict__ bnsh,
    const float* __restrict__ Res, long rBS, int M, int N, int K) {
  __shared__ float As[64][36];   // row stride 144B (16B aligned, conflict-free)
  __shared__ float Bs[32][132];  // row stride 528B
  int bz = blockIdx.z;
  int bb = bz >> halfShift;
  long half = (long)(bz & ((1 << halfShift) - 1));
  const float* Ab = A + (long)bz * aBS;
  const float* Bb = Bm + (long)bb * bBS;
  long oBase = (long)bb * oBS + half * oHalfOff;
  long rBase = (long)bb * rBS + half * oHalfOff;
  int tid = threadIdx.x;
  int mBase = blockIdx.y * 64;
  int nBase = blockIdx.x * 128;
  int lane = tid & 31;
  int wave = tid >> 5;
  int wm = wave & 3, wn = wave >> 2;   // 4 M-waves x 2 N-waves
  int m0 = wm * 16, nW = wn * 64;
  int ml = lane & 15, kg = lane >> 4;  // ISA 7.12.2 fragment layout

  // staging maps: A = 8 quads x 32 rows (x2); B = 32 quads x 8 rows (x4)
  int aq = tid & 7, ar = tid >> 3;
  int bq = tid & 31, br = tid >> 5;
  const float* aP0 = Ab + (long)(mBase + ar) * K + aq * 4;
  const float* aP1 = aP0 + (long)32 * K;
  const float* bP = Bb + (long)br * ldb + nBase + bq * 4;
  bool aok0 = (mBase + ar) < M;
  bool aok1 = (mBase + ar + 32) < M;
  const f4 f4z = {0.f, 0.f, 0.f, 0.f};

  v8f acc[4];
#pragma unroll
  for (int s = 0; s < 4; ++s) acc[s] = (v8f){0.f, 0.f, 0.f, 0.f,
                                             0.f, 0.f, 0.f, 0.f};
  // prologue: load chunk 0 into registers (batched b128 loads)
  f4 va0 = aok0 ? *(const f4*)aP0 : f4z;
  f4 va1 = aok1 ? *(const f4*)aP1 : f4z;
  f4 vb0 = *(const f4*)(bP);
  f4 vb1 = *(const f4*)(bP + (long)8 * ldb);
  f4 vb2 = *(const f4*)(bP + (long)16 * ldb);
  f4 vb3 = *(const f4*)(bP + (long)24 * ldb);

  for (int k0 = 0; k0 < K; k0 += 32) {
    *(f4*)&As[ar][aq * 4] = va0;
    *(f4*)&As[ar + 32][aq * 4] = va1;
    *(f4*)&Bs[br][bq * 4] = vb0;
    *(f4*)&Bs[br + 8][bq * 4] = vb1;
    *(f4*)&Bs[br + 16][bq * 4] = vb2;
    *(f4*)&Bs[br + 24][bq * 4] = vb3;
    __syncthreads();
    aP0 += 32;
    aP1 += 32;
    bP += (long)32 * ldb;
    if (k0 + 32 < K) {  // issue next chunk's loads before the compute section
      va0 = aok0 ? *(const f4*)aP0 : f4z;
      va1 = aok1 ? *(const f4*)aP1 : f4z;
      vb0 = *(const f4*)(bP);
      vb1 = *(const f4*)(bP + (long)8 * ldb);
      vb2 = *(const f4*)(bP + (long)16 * ldb);
      vb3 = *(const f4*)(bP + (long)24 * ldb);
      if (k0 + 64 < K) {  // hint L2 for the streamed operand two chunks out
        __builtin_prefetch(bP + (long)32 * ldb, 0, 0);
        __builtin_prefetch(bP + (long)48 * ldb, 0, 0);
      }
    }
#pragma unroll
    for (int kk = 0; kk < 32; kk += 4) {
      v2f a, b0, b1, b2, b3;
      a.x = As[m0 + ml][kk + 2 * kg];
      a.y = As[m0 + ml][kk + 2 * kg + 1];
      b0.x = Bs[kk + 2 * kg][nW + ml];
      b0.y = Bs[kk + 2 * kg + 1][nW + ml];
      b1.x = Bs[kk + 2 * kg][nW + 16 + ml];
      b1.y = Bs[kk + 2 * kg + 1][nW + 16 + ml];
      b2.x = Bs[kk + 2 * kg][nW + 32 + ml];
      b2.y = Bs[kk + 2 * kg + 1][nW + 32 + ml];
      b3.x = Bs[kk + 2 * kg][nW + 48 + ml];
      b3.y = Bs[kk + 2 * kg + 1][nW + 48 + ml];
      acc[0] = wmma_f32_16x16x4(a, b0, acc[0]);
      acc[1] = wmma_f32_16x16x4(a, b1, acc[1]);
      acc[2] = wmma_f32_16x16x4(a, b2, acc[2]);
      acc[3] = wmma_f32_16x16x4(a, b3, acc[3]);
    }
    __syncthreads();
  }
  // epilogue; C/D layout: VGPR j -> rows j (lanes 0-15) and j+8 (lanes 16-31)
#pragma unroll
  for (int j = 0; j < 8; ++j) {
    int gm = mBase + m0 + j + 8 * kg;
    if (gm >= M) continue;
    float bv = bias[gm];
    float s = 1.f, t = 0.f;
    if (EPI != EPI_BIAS) {
      s = bnsc[gm];
      t = bnsh[gm];
    }
#pragma unroll
    for (int ss = 0; ss < 4; ++ss) {
      int col = nBase + nW + 16 * ss + ml;
      float v = acc[ss][j] + bv;
      if (EPI == EPI_GELU_BN) v = gelu_erf(v);
      if (EPI != EPI_BIAS) v = v * s + t;
      if (EPI == EPI_BN_RES) v += Res[rBase + (long)gm * ldo + col];
      Out[oBase + (long)gm * ldo + col] = v;
    }
  }
}

// ============================ host launcher ============================
extern "C" void kernel_launch(void* const* d_in, const int* in_sizes, int n_in,
                              void* d_out, int out_size, void* d_ws,
                              size_t ws_size, hipStream_t stream) {
  (void)in_sizes; (void)n_in; (void)out_size;
  const int B = 16, CH = 256, CL = 64, HID = 1024, H = 96, W = 32;
  const long N = 3072, N2 = 1536;

  const float* X = (const float*)d_in[0];
  const float* X0 = (const float*)d_in[1];
  const float* lpu_h_w = (const float*)d_in[2];
  const float* lpu_h_b = (const float*)d_in[3];
  const float* lpu_l_w = (const float*)d_in[4];
  const float* lpu_l_b = (const float*)d_in[5];
  const float* n1_g = (const float*)d_in[6];
  const float* n1_b = (const float*)d_in[7];
  const float* n2_g = (const float*)d_in[8];
  const float* n2_b = (const float*)d_in[9];
  const float* n3_g = (const float*)d_in[10];
  const float* n3_b = (const float*)d_in[11];
  const float* cnl_g_w = (const float*)d_in[12];
  const float* cnl_g_b = (const float*)d_in[13];
  const float* cnl_th_w = (const float*)d_in[14];
  const float* cnl_th_b = (const float*)d_in[15];
  const float* cnl_ph_w = (const float*)d_in[16];
  const float* cnl_ph_b = (const float*)d_in[17];
  const float* cnl_W_w = (const float*)d_in[18];
  const float* cnl_W_b = (const float*)d_in[19];
  const float* pnl_g_w = (const float*)d_in[20];
  const float* pnl_g_b = (const float*)d_in[21];
  const float* pnl_th_w = (const float*)d_in[22];
  const float* pnl_th_b = (const float*)d_in[23];
  const float* pnl_ph_w = (const float*)d_in[24];
  const float* pnl_ph_b = (const float*)d_in[25];
  const float* pnl_W_w = (const float*)d_in[26];
  const float* pnl_W_b = (const float*)d_in[27];
  const float* ff_c1_w = (const float*)d_in[28];
  const float* ff_c1_b = (const float*)d_in[29];
  const float* ff_dw_w = (const float*)d_in[30];
  const float* ff_dw_b = (const float*)d_in[31];
  const float* ff_c2_w = (const float*)d_in[32];
  const float* ff_c2_b = (const float*)d_in[33];

  float* ws = (float*)d_ws;
  const long SZ_BIG = (long)B * CH * N;   // 12,582,912
  const long SZ_HID = (long)B * HID * N;  // 50,331,648
  const long SZ_LOW = (long)B * CL * N;   // 3,145,728
  long off = 0;
  float* buf_xh_z = ws + off; off += SZ_BIG;  // xh, later z
  float* buf_xn = ws + off; off += SZ_BIG;    // xn, later zn
  float* buf_z2 = ws + off; off += SZ_BIG;
  float* buf_h1 = ws + off; off += SZ_HID;
  float* buf_h2 = ws + off; off += SZ_HID;
  float* buf_xl_tx = ws + off; off += SZ_LOW;   // xl, later tx
  float* buf_x0n = ws + off; off += SZ_LOW;
  float* buf_th_px = ws + off; off += SZ_LOW;   // th, later px
  float* buf_ph_gx2 = ws + off; off += SZ_LOW;  // ph, later gx2
  float* buf_gxb = ws + off; off += SZ_LOW;
  float* buf_E = ws + off; off += (long)B * 64 * 64;
  float* buf_M2 = ws + off; off += (long)B * 256 * 64;
  float* buf_S = ws + off; off += (long)B * 64 * 64;
  float* buf_Wf = ws + off; off += (long)B * 2 * 256 * 64;
  float* bn_cnl_sc = ws + off; off += 256;
  float* bn_cnl_sh = ws + off; off += 256;
  float* bn_pnl_sc = ws + off; off += 256;
  float* bn_pnl_sh = ws + off; off += 256;
  float* bn_ff1_sc = ws + off; off += 1024;
  float* bn_ff1_sh = ws + off; off += 1024;
  float* bn_ff2_sc = ws + off; off += 1024;
  float* bn_ff2_sh = ws + off; off += 1024;
  float* bn_ff3_sc = ws + off; off += 256;
  float* bn_ff3_sh = ws + off; off += 256;
  if ((size_t)off * sizeof(float) > ws_size) return;  // insufficient scratch

  dim3 blk(256);
  // BN prep (param order in d_in: g,b,m,v)
  bn_prep_kernel<<<1, 256, 0, stream>>>((const float*)d_in[34], (const float*)d_in[35],
                                        (const float*)d_in[36], (const float*)d_in[37],
                                        bn_cnl_sc, bn_cnl_sh, 256);
  bn_prep_kernel<<<1, 256, 0, stream>>>((const float*)d_in[38], (const float*)d_in[39],
                                        (const float*)d_in[40], (const float*)d_in[41],
                                        bn_pnl_sc, bn_pnl_sh, 256);
  bn_prep_kernel<<<4, 256, 0, stream>>>((const float*)d_in[42], (const float*)d_in[43],
                                        (const float*)d_in[44], (const float*)d_in[45],
                                        bn_ff1_sc, bn_ff1_sh, 1024);
  bn_prep_kernel<<<4, 256, 0, stream>>>((const float*)d_in[46], (const float*)d_in[47],
                                        (const float*)d_in[48], (const float*)d_in[49],
                                        bn_ff2_sc, bn_ff2_sh, 1024);
  bn_prep_kernel<<<1, 256, 0, stream>>>((const float*)d_in[50], (const float*)d_in[51],
                                        (const float*)d_in[52], (const float*)d_in[53],
                                        bn_ff3_sc, bn_ff3_sh, 256);

  // Stage 1: LPU depthwise + residual
  dwconv3x3_kernel<0><<<dim3(H / 8, CH, B), blk, 0, stream>>>(
      X, lpu_h_w, lpu_h_b, nullptr, nullptr, buf_xh_z, CH, H, W);
  dwconv3x3_kernel<0><<<dim3(H / 8, CL, B), blk, 0, stream>>>(
      X0, lpu_l_w, lpu_l_b, nullptr, nullptr, buf_xl_tx, CL, H, W);

  // Stage 2: channel LayerNorms
  ln_chan_kernel<<<(B * (int)N) / 256, blk, 0, stream>>>(buf_xh_z, n1_g, n1_b,
                                                         buf_xn, CH, (int)N);
  ln_chan_kernel<<<(B * (int)N) / 256, blk, 0, stream>>>(buf_xl_tx, n2_g, n2_b,
                                                         buf_x0n, CL, (int)N);

  // Stage 3: CNL projections (theta from xn; phi, g from x0n)
  gemm_wmma_kernel<EPI_BIAS><<<dim3(24, 1, B), blk, 0, stream>>>(
      cnl_th_w, 0, buf_xn, (long)CH * N, (int)N, buf_th_px, (long)CL * N,
      (int)N, 0, 0, cnl_th_b, nullptr, nullptr, nullptr, 0, CL, (int)N, CH);
  gemm_wmma_kernel<EPI_BIAS><<<dim3(24, 1, B), blk, 0, stream>>>(
      cnl_ph_w, 0, buf_x0n, (long)CL * N, (int)N, buf_ph_gx2, (long)CL * N,
      (int)N, 0, 0, cnl_ph_b, nullptr, nullptr, nullptr, 0, CL, (int)N, CL);
  gemm_wmma_kernel<EPI_BIAS><<<dim3(24, 1, B), blk, 0, stream>>>(
      cnl_g_w, 0, buf_x0n, (long)CL * N, (int)N, buf_gxb, (long)CL * N, (int)N,
      0, 0, cnl_g_b, nullptr, nullptr, nullptr, 0, CL, (int)N, CL);

  // Stage 4: E = theta @ phi^T ; M2 = (Wc @ E)/64
  abt64_kernel<<<B, blk, 0, stream>>>(buf_th_px, buf_ph_gx2, buf_E, (int)N,
                                      (long)CL * N, (long)CL * N);
  fold_cnl_kernel<<<dim3(64, B), blk, 0, stream>>>(cnl_W_w, buf_E, buf_M2);

  // Stage 5: z = BN(M2 @ g + bc) + xn   (writes into xh slot)
  gemm_wmma_kernel<EPI_BN_RES><<<dim3(24, 4, B), blk, 0, stream>>>(
      buf_M2, (long)256 * 64, buf_gxb, (long)CL * N, (int)N, buf_xh_z,
      (long)CH * N, (int)N, 0, 0, cnl_W_b, bn_cnl_sc, bn_cnl_sh, buf_xn,
      (long)CH * N, CH, (int)N, CL);

  // Stage 6: PNL projections (theta from z; phi, g from x0n), RED=32 channels
  gemm_wmma_kernel<EPI_BIAS><<<dim3(24, 1, B), blk, 0, stream>>>(
      pnl_th_w, 0, buf_xh_z, (long)CH * N, (int)N, buf_xl_tx, (long)32 * N,
      (int)N, 0, 0, pnl_th_b, nullptr, nullptr, nullptr, 0, 32, (int)N, CH);
  gemm_wmma_kernel<EPI_BIAS><<<dim3(24, 1, B), blk, 0, stream>>>(
      pnl_ph_w, 0, buf_x0n, (long)CL * N, (int)N, buf_th_px, (long)32 * N,
      (int)N, 0, 0, pnl_ph_b, nullptr, nullptr, nullptr, 0, 32, (int)N, CL);
  gemm_wmma_kernel<EPI_BIAS><<<dim3(24, 1, B), blk, 0, stream>>>(
      pnl_g_w, 0, buf_x0n, (long)CL * N, (int)N, buf_ph_gx2, (long)32 * N,
      (int)N, 0, 0, pnl_g_b, nullptr, nullptr, nullptr, 0, 32, (int)N, CL);

  // Stage 7: S = px64 @ gx64^T  (flat [32,N] reinterpreted as [64, N/2])
  abt64_kernel<<<B, blk, 0, stream>>>(buf_th_px, buf_ph_gx2, buf_S, (int)N2,
                                      (long)32 * N, (long)32 * N);
  // Stage 8a: fold pnl_W through S -> WE/WO per batch
  fold_pnl_kernel<<<dim3(128, B), blk, 0, stream>>>(pnl_W_w, buf_S, buf_Wf);
  // Stage 8b: z2 = BN(Wf[half] @ tx64 + bp) + z   (half-mode GEMM, grid.z=32)
  gemm_wmma_kernel<EPI_BN_RES><<<dim3(12, 4, 2 * B), blk, 0, stream>>>(
      buf_Wf, (long)256 * 64, buf_xl_tx, (long)32 * N, (int)N2, buf_z2,
      (long)CH * N, (int)N, N2, 1, pnl_W_b, bn_pnl_sc, bn_pnl_sh, buf_xh_z,
      (long)CH * N, CH, (int)N2, CL);

  // Stage 9: zn = LN(z2)  (into xn slot)
  ln_chan_kernel<<<(B * (int)N) / 256, blk, 0, stream>>>(buf_z2, n3_g, n3_b,
                                                         buf_xn, CH, (int)N);
  // Stage 10: h1 = BN1(gelu(Wc1 @ zn + b1))
  gemm_wmma_kernel<EPI_GELU_BN><<<dim3(24, 16, B), blk, 0, stream>>>(
      ff_c1_w, 0, buf_xn, (long)CH * N, (int)N, buf_h1, (long)HID * N, (int)N,
      0, 0, ff_c1_b, bn_ff1_sc, bn_ff1_sh, nullptr, 0, HID, (int)N, CH);
  // Stage 11: h2 = h1 + gelu(BN2(dw3x3(h1) + bdw))
  dwconv3x3_kernel<1><<<dim3(H / 8, HID, B), blk, 0, stream>>>(
      buf_h1, ff_dw_w, ff_dw_b, bn_ff2_sc, bn_ff2_sh, buf_h2, HID, H, W);
  // Stage 12: out = z2 + BN3(Wc2 @ h2 + b2)
  gemm_wmma_kernel<EPI_BN_RES><<<dim3(24, 4, B), blk, 0, stream>>>(
      ff_c2_w, 0, buf_h2, (long)HID * N, (int)N, (float*)d_out, (long)CH * N,
      (int)N, 0, 0, ff_c2_b, bn_ff3_sc, bn_ff3_sh, buf_z2, (long)CH * N, CH,
      (int)N, HID);
}